// SimpleGNN_94489281042
// MI455X (gfx1250) — compile-verified
//
#include <hip/hip_runtime.h>
#include <cstdint>
#include <cstddef>

// ---------------------------------------------------------------------------
// SimpleGNN (2-layer GCN) for MI455X / gfx1250, wave32.
// Memory-bound workload: ~1.3 GB of gather+atomic traffic dominates (~60us at
// 23.3 TB/s); GEMMs are ~2.5 GFLOP total -> use native f32 WMMA 16x16x4 to
// match reference precision exactly, aggregation via fp32 global atomics.
// ---------------------------------------------------------------------------

typedef __attribute__((ext_vector_type(2))) float v2f;
typedef __attribute__((ext_vector_type(8))) float v8f;

#define IN_CH  128
#define HID_CH 128
#define OUT_CH 64

__device__ __forceinline__ void atomAddF32(float* p, float v) {
    // relaxed, device scope -> lowers to global_atomic_add_f32 (no CAS loop)
    __hip_atomic_fetch_add(p, v, __ATOMIC_RELAXED, __HIP_MEMORY_SCOPE_AGENT);
}

// --- degree / normalization ------------------------------------------------

__global__ void k_init_deg(float* __restrict__ deg, int n) {
    int i = blockIdx.x * blockDim.x + threadIdx.x;
    if (i < n) deg[i] = 1.0f;  // self-loop contributes 1
}

__global__ void k_deg_edges(const int* __restrict__ dst, float* __restrict__ deg, int E) {
    int e = blockIdx.x * blockDim.x + threadIdx.x;
    if (e < E) atomAddF32(&deg[dst[e]], 1.0f);
}

__global__ void k_dinv(float* __restrict__ deg, int n) {
    int i = blockIdx.x * blockDim.x + threadIdx.x;
    if (i < n) deg[i] = rsqrtf(deg[i]);  // deg >= 1 always (self-loops)
}

__global__ void k_norm(const int* __restrict__ src, const int* __restrict__ dst,
                       const float* __restrict__ dinv, float* __restrict__ norm, int E) {
    int e = blockIdx.x * blockDim.x + threadIdx.x;
    if (e < E) norm[e] = dinv[src[e]] * dinv[dst[e]];
}

// --- dense GEMM: C[M,N] = A[M,K] * B[K,N] via V_WMMA_F32_16X16X4_F32 --------
// One wave per 16x16 output tile. Fragment layouts per CDNA5 ISA 7.12.2:
//   A 16x4 f32 : lane(l,half): VGPR0 = A[m][k+2*half], VGPR1 = A[m][k+2*half+1]
//   B 4x16 f32 : VGPR0 = B[k+2*half][n], VGPR1 = B[k+2*half+1][n]
//   C/D 16x16  : VGPR r = C[m0 + r + 8*half][n0 + l]

__global__ __launch_bounds__(32)
void k_gemm_wmma_f32(const float* __restrict__ A, const float* __restrict__ B,
                     float* __restrict__ C, int M, int N, int K) {
    const int lane = threadIdx.x & 31;
    const int half = lane >> 4;     // 0 or 1
    const int l    = lane & 15;
    const int m0   = blockIdx.x * 16;
    const int n0   = blockIdx.y * 16;

    int ra = m0 + l;
    if (ra > M - 1) ra = M - 1;           // clamp tail reads; keep EXEC all-1s
    const float* arow = A + (size_t)ra * K;

    v8f acc = {};
    for (int k = 0; k < K; k += 4) {
        const int ka = k + 2 * half;
        v2f a = *(const v2f*)(arow + ka);                 // contiguous K pair
        v2f b;
        b.x = B[(size_t)ka * N + n0 + l];
        b.y = B[(size_t)(ka + 1) * N + n0 + l];
        acc = __builtin_amdgcn_wmma_f32_16x16x4_f32(
            /*neg_a=*/false, a, /*neg_b=*/false, b,
            /*c_mod=*/(short)0, acc, /*reuse_a=*/false, /*reuse_b=*/false);
    }

#pragma unroll
    for (int r = 0; r < 8; ++r) {
        int rm = m0 + r + 8 * half;
        if (rm < M) C[(size_t)rm * N + (n0 + l)] = acc[r];
    }
}

// --- aggregation -----------------------------------------------------------
// agg[i] = h[i] * dinv[i]^2 + bias   (self-loop message + bias folded in)
template <int C>
__global__ void k_init_agg(const float* __restrict__ h, const float* __restrict__ dinv,
                           const float* __restrict__ bias, float* __restrict__ agg, int n) {
    const int lanes = C / 4;
    long long t = (long long)blockIdx.x * blockDim.x + threadIdx.x;
    int i  = (int)(t / lanes);
    int c4 = (int)(t % lanes) * 4;
    if (i >= n) return;
    float w = dinv[i];
    w *= w;
    float4 v = *(const float4*)(h + (size_t)i * C + c4);
    float4 b = *(const float4*)(bias + c4);
    float4 o = make_float4(v.x * w + b.x, v.y * w + b.y, v.z * w + b.z, v.w * w + b.w);
    *(float4*)(agg + (size_t)i * C + c4) = o;
}

// agg[dst[e]] += h[src[e]] * norm[e]   (C/4 lanes per edge, float4 gather)
template <int C>
__global__ void k_edge_agg(const int* __restrict__ src, const int* __restrict__ dst,
                           const float* __restrict__ norm, const float* __restrict__ h,
                           float* __restrict__ agg, int E) {
    const int lanes = C / 4;
    long long t = (long long)blockIdx.x * blockDim.x + threadIdx.x;
    int e  = (int)(t / lanes);
    int c4 = (int)(t % lanes) * 4;
    if (e >= E) return;
    const int s  = src[e];
    const int d  = dst[e];
    const float nv = norm[e];
    float4 v = *(const float4*)(h + (size_t)s * C + c4);
    float* a = agg + (size_t)d * C + c4;
    atomAddF32(a + 0, v.x * nv);
    atomAddF32(a + 1, v.y * nv);
    atomAddF32(a + 2, v.z * nv);
    atomAddF32(a + 3, v.w * nv);
}

__global__ void k_relu(float* __restrict__ x, long long n) {
    long long i = (long long)blockIdx.x * blockDim.x + threadIdx.x;
    if (i < n) x[i] = fmaxf(x[i], 0.0f);
}

// ---------------------------------------------------------------------------

static inline size_t alignup(size_t v) { return (v + 255) & ~(size_t)255; }
static inline int cdiv(long long a, long long b) { return (int)((a + b - 1) / b); }

extern "C" void kernel_launch(void* const* d_in, const int* in_sizes, int n_in,
                              void* d_out, int out_size, void* d_ws, size_t ws_size,
                              hipStream_t stream) {
    const float* x  = (const float*)d_in[0];
    const int*   ei = (const int*)d_in[1];   // edge_index [2, E] (int32)
    const float* W1 = (const float*)d_in[2];
    const float* b1 = (const float*)d_in[3];
    const float* W2 = (const float*)d_in[4];
    const float* b2 = (const float*)d_in[5];
    float* out = (float*)d_out;

    const int Nn = in_sizes[0] / IN_CH;      // 50000
    const int E  = in_sizes[1] / 2;          // 800000
    const int* src = ei;
    const int* dst = ei + E;

    // workspace carve-out
    char*  ws  = (char*)d_ws;
    size_t off = 0;
    float* dinv = (float*)(ws + off); off += alignup((size_t)Nn * 4);
    float* nrm  = (float*)(ws + off); off += alignup((size_t)E * 4);
    float* h1   = (float*)(ws + off); off += alignup((size_t)Nn * HID_CH * 4);
    float* agg1 = (float*)(ws + off); off += alignup((size_t)Nn * HID_CH * 4);
    float* h2   = (float*)(ws + off); off += alignup((size_t)Nn * OUT_CH * 4);
    (void)ws_size;

    const int TB = 256;

    // 1) degrees (self-loop = 1) + dinv + per-edge norms (shared by both layers)
    k_init_deg<<<cdiv(Nn, TB), TB, 0, stream>>>(dinv, Nn);
    k_deg_edges<<<cdiv(E, TB), TB, 0, stream>>>(dst, dinv, E);
    k_dinv<<<cdiv(Nn, TB), TB, 0, stream>>>(dinv, Nn);
    k_norm<<<cdiv(E, TB), TB, 0, stream>>>(src, dst, dinv, nrm, E);

    // 2) layer 1: h1 = x @ W1  (WMMA f32), then normalized aggregation + ReLU
    {
        dim3 grid(cdiv(Nn, 16), HID_CH / 16);
        k_gemm_wmma_f32<<<grid, 32, 0, stream>>>(x, W1, h1, Nn, HID_CH, IN_CH);
    }
    k_init_agg<HID_CH><<<cdiv((long long)Nn * (HID_CH / 4), TB), TB, 0, stream>>>(h1, dinv, b1, agg1, Nn);
    k_edge_agg<HID_CH><<<cdiv((long long)E * (HID_CH / 4), TB), TB, 0, stream>>>(src, dst, nrm, h1, agg1, E);
    k_relu<<<cdiv((long long)Nn * HID_CH, TB), TB, 0, stream>>>(agg1, (long long)Nn * HID_CH);

    // 3) layer 2: h2 = relu(agg1) @ W2, aggregate straight into d_out
    {
        dim3 grid(cdiv(Nn, 16), OUT_CH / 16);
        k_gemm_wmma_f32<<<grid, 32, 0, stream>>>(agg1, W2, h2, Nn, OUT_CH, HID_CH);
    }
    k_init_agg<OUT_CH><<<cdiv((long long)Nn * (OUT_CH / 4), TB), TB, 0, stream>>>(h2, dinv, b2, out, Nn);
    k_edge_agg<OUT_CH><<<cdiv((long long)E * (OUT_CH / 4), TB), TB, 0, stream>>>(src, dst, nrm, h2, out, E);
}